// SignalAwareMVNProfile_78314433675743
// MI455X (gfx1250) — compile-verified
//
#include <hip/hip_runtime.h>
#include <hip/hip_bf16.h>
#include <cstdint>

// ---------------------------------------------------------------------------
// SignalAwareMVNProfile for MI455X (gfx1250, wave32)
//   B = 32768, DMODEL = 1024, NPIX = 3*21*21 = 1323
//   Stage 1: pack [W_mean | W_scale | 0-pad] into WMMA B-operand layout
//   Stage 2: fp32 WMMA GEMM (v_wmma_f32_16x16x4_f32) -> per-batch 9 params
//   Stage 3: per-batch wave32 kernel: 3x3 Cholesky + softmax(-0.5*maha)
// ---------------------------------------------------------------------------

#define DMODEL 1024
#define NBATCH 32768
#define NPIX   1323      // 3*21*21
#define NITER  42        // ceil(1323/32)
#define KSTEPS (DMODEL / 4)

typedef __attribute__((ext_vector_type(2))) float v2f;
typedef __attribute__((ext_vector_type(8))) float v8f;

// ---------------------------------------------------------------------------
// Kernel 1: pack weights into the V_WMMA_F32_16X16X4_F32 B-matrix lane layout.
// B operand is 4x16 fp32 across 2 VGPRs of a wave32:
//   VGPR0: lanes 0-15 -> K=0, lanes 16-31 -> K=2 ; VGPR1: K=1 / K=3 ; N = lane&15
// We store it so lane L reads a contiguous float2 at Bp[step*64 + L*2].
// Logical W[k][n]: n<3 -> W_mean[k][n], 3<=n<9 -> W_scale[k][n-3], else 0.
// ---------------------------------------------------------------------------
__global__ void mvn_pack_w(const float* __restrict__ Wm,
                           const float* __restrict__ Ws,
                           float* __restrict__ Bp) {
    int idx = blockIdx.x * blockDim.x + threadIdx.x;   // 0 .. 256*64-1
    if (idx >= KSTEPS * 64) return;
    int s    = idx >> 6;        // K-step
    int r    = idx & 63;        // r = lane*2 + v
    int lane = r >> 1;
    int v    = r & 1;
    int n    = lane & 15;
    int k    = s * 4 + v + 2 * (lane >> 4);
    float val = 0.0f;
    if (n < 3)      val = Wm[k * 3 + n];
    else if (n < 9) val = Ws[k * 6 + (n - 3)];
    Bp[idx] = val;
}

// ---------------------------------------------------------------------------
// Kernel 2: GEMM rep(32768x1024) x W(1024x16-padded) via fp32 WMMA.
// One wave per 16 batch rows; 256 k-steps of v_wmma_f32_16x16x4_f32.
// A operand layout (32-bit A 16x4): lane L -> row M=L&15, K = 2*(L>>4)+{0,1}.
// C/D layout: lane L -> N=L&15, VGPR r -> M = r + 8*(L>>4).
// Result (+bias) goes to params[b*16 + n], n=0..8.
// ---------------------------------------------------------------------------
__global__ void mvn_gemm_params(const float* __restrict__ rep,
                                const float* __restrict__ Bp,
                                const float* __restrict__ b_mean,
                                const float* __restrict__ b_scale,
                                float* __restrict__ params) {
    const int lane  = threadIdx.x & 31;
    const int wave  = threadIdx.x >> 5;
    const int tile  = blockIdx.x * (blockDim.x >> 5) + wave;   // 16-row tile
    const int khalf = lane >> 4;
    const int row   = tile * 16 + (lane & 15);

    const float* __restrict__ arow = rep + (size_t)row * DMODEL + 2 * khalf;
    const float* __restrict__ brow = Bp + lane * 2;

    v8f acc = {0.f, 0.f, 0.f, 0.f, 0.f, 0.f, 0.f, 0.f};

#pragma unroll 4
    for (int s = 0; s < KSTEPS; ++s) {
        v2f a = *(const v2f*)(arow + s * 4);
        v2f b = *(const v2f*)(brow + s * 64);
        // (neg_a, A, neg_b, B, c_mod, C, reuse_a, reuse_b)
        acc = __builtin_amdgcn_wmma_f32_16x16x4_f32(
            false, a, false, b, (short)0, acc, false, false);
    }

    const int n = lane & 15;
    if (n < 9) {
        const float bias = (n < 3) ? b_mean[n] : b_scale[n - 3];
        float* __restrict__ prow = params + (size_t)tile * 16 * 16;
#pragma unroll
        for (int r = 0; r < 8; ++r) {
            int m = r + 8 * khalf;
            prow[m * 16 + n] = acc[r] + bias;
        }
    }
}

// ---------------------------------------------------------------------------
// Kernel 3: one wave32 per batch element.
//   - per-batch 3x3 setup computed redundantly by all lanes (wave uniform)
//   - softmax constants (-0.5*logdet - 1.5*log2pi) cancel: prof=softmax(-maha/2)
//   - 42 pixels per lane kept in registers (fixed unrolled trip count)
// ---------------------------------------------------------------------------
__device__ __forceinline__ float softplus_f(float x) {
    return (x > 15.0f) ? x : log1pf(expf(x));
}
__device__ __forceinline__ float elu1_f(float x) {   // elu(x) + 1
    return (x > 0.0f) ? (x + 1.0f) : expf(x);
}

__global__ void mvn_profile(const float* __restrict__ params,
                            const float* __restrict__ signal_prob,
                            const float* __restrict__ log_base_sigma,
                            float* __restrict__ out) {
    const int lane = threadIdx.x & 31;
    const int wave = threadIdx.x >> 5;
    const int b    = blockIdx.x * (blockDim.x >> 5) + wave;

    const float* __restrict__ p = params + (size_t)b * 16;
    const float m0 = p[0], m1 = p[1], m2 = p[2];

    const float e0 = elu1_f(p[3]);
    const float e1 = elu1_f(p[4]);
    const float e2 = elu1_f(p[5]);
    const float e3 = elu1_f(p[6]);
    const float e4 = elu1_f(p[7]);
    const float e5 = elu1_f(p[8]);

    const float l00 = softplus_f(e0);
    const float l10 = e1;
    const float l11 = softplus_f(e2);
    const float l20 = e3;
    const float l21 = e4;
    const float l22 = softplus_f(e5);

    // predicted_cov = L L^T
    float c00 = l00 * l00;
    float c01 = l00 * l10;
    float c02 = l00 * l20;
    float c11 = l10 * l10 + l11 * l11;
    float c12 = l10 * l20 + l11 * l21;
    float c22 = l20 * l20 + l21 * l21 + l22 * l22;

    // blend with base_sigma^2 * I
    float sp = signal_prob[b];
    sp = fminf(fmaxf(sp, 0.0f), 1.0f);
    const float a   = sp * sp * sp;
    const float bs  = expf(log_base_sigma[0]);
    const float oma = (1.0f - a) * bs * bs;
    c00 = a * c00 + oma;  c11 = a * c11 + oma;  c22 = a * c22 + oma;
    c01 *= a;  c02 *= a;  c12 *= a;

    // 3x3 Cholesky, keep inverse diagonals
    const float t00 = sqrtf(c00);
    const float i00 = 1.0f / t00;
    const float t10 = c01 * i00;
    const float t20 = c02 * i00;
    const float t11 = sqrtf(c11 - t10 * t10);
    const float i11 = 1.0f / t11;
    const float t21 = (c12 - t10 * t20) * i11;
    const float t22 = sqrtf(c22 - t20 * t20 - t21 * t21);
    const float i22 = 1.0f / t22;

    // pass 1: mahalanobis per pixel, track wave min
    float maha[NITER];
    float mn = 3.4e38f;
#pragma unroll
    for (int i = 0; i < NITER; ++i) {
        int pix = i * 32 + lane;
        int pc  = (pix < NPIX) ? pix : (NPIX - 1);   // clamp; dup is harmless for min
        int z   = pc / 441;
        int rem = pc - z * 441;
        int y   = rem / 21;
        int x   = rem - y * 21;
        float d0 = (float)(x - 10) - m0;
        float d1 = (float)(y - 10) - m1;
        float d2 = (float)(z - 1)  - m2;
        float y0 = d0 * i00;
        float y1 = (d1 - t10 * y0) * i11;
        float y2 = (d2 - t20 * y0 - t21 * y1) * i22;
        float mh = y0 * y0 + y1 * y1 + y2 * y2;
        maha[i] = mh;
        mn = fminf(mn, mh);
    }
#pragma unroll
    for (int off = 16; off > 0; off >>= 1)
        mn = fminf(mn, __shfl_xor(mn, off, 32));

    // pass 2: exp and wave sum (softmax constants cancelled)
    float sum = 0.0f;
#pragma unroll
    for (int i = 0; i < NITER; ++i) {
        int pix = i * 32 + lane;
        float e = __expf(-0.5f * (maha[i] - mn));
        maha[i] = e;
        if (pix < NPIX) sum += e;
    }
#pragma unroll
    for (int off = 16; off > 0; off >>= 1)
        sum += __shfl_xor(sum, off, 32);
    const float inv = 1.0f / (sum + 1e-10f);

    // pass 3: coalesced normalized write
    float* __restrict__ orow = out + (size_t)b * NPIX;
#pragma unroll
    for (int i = 0; i < NITER; ++i) {
        int pix = i * 32 + lane;
        if (pix < NPIX) orow[pix] = maha[i] * inv;
    }
}

// ---------------------------------------------------------------------------
// Launcher. Workspace layout: [0, 64KB) packed B operand,
//                             [64KB, 64KB + 2MB) per-batch params (B x 16 f32).
// ---------------------------------------------------------------------------
extern "C" void kernel_launch(void* const* d_in, const int* in_sizes, int n_in,
                              void* d_out, int out_size, void* d_ws, size_t ws_size,
                              hipStream_t stream) {
    const float* rep      = (const float*)d_in[0];   // (B, 1024)
    const float* sig      = (const float*)d_in[1];   // (B,)
    const float* W_mean   = (const float*)d_in[2];   // (1024, 3)
    const float* b_mean   = (const float*)d_in[3];   // (3,)
    const float* W_scale  = (const float*)d_in[4];   // (1024, 6)
    const float* b_scale  = (const float*)d_in[5];   // (6,)
    const float* log_bs   = (const float*)d_in[6];   // scalar
    float*       out      = (float*)d_out;           // (B, 1323)

    float* Bp     = (float*)d_ws;                              // 16384 floats
    float* params = (float*)((char*)d_ws + 64 * 1024);         // B*16 floats

    // Stage 1: pack weights (16384 elements)
    mvn_pack_w<<<(KSTEPS * 64 + 255) / 256, 256, 0, stream>>>(W_mean, W_scale, Bp);

    // Stage 2: WMMA GEMM. 2048 tiles of 16 rows, 8 waves per 256-thread block.
    mvn_gemm_params<<<(NBATCH / 16) / 8, 256, 0, stream>>>(rep, Bp, b_mean, b_scale, params);

    // Stage 3: one wave per batch, 8 waves per block.
    mvn_profile<<<NBATCH / 8, 256, 0, stream>>>(params, sig, log_bs, out);
}